// GCN3Layer_44212393345738
// MI455X (gfx1250) — compile-verified
//
#include <hip/hip_runtime.h>
#include <hip/hip_bf16.h>

typedef float v2f __attribute__((ext_vector_type(2)));
typedef float v8f __attribute__((ext_vector_type(8)));

#define GCN_N_NODES 100000
#define GCN_N_EDGES 1600000

// ---------------------------------------------------------------------------
// Degree / normalization
// ---------------------------------------------------------------------------
__global__ void gcn_init_deg_kernel(float* __restrict__ deg, int n) {
    int i = blockIdx.x * blockDim.x + threadIdx.x;
    if (i < n) deg[i] = 1.0f;  // self-loop contributes 1 to each node's degree
}

__global__ void gcn_edge_deg_kernel(const int* __restrict__ dst,
                                    float* __restrict__ deg, int nedges) {
    int e = blockIdx.x * blockDim.x + threadIdx.x;
    if (e < nedges) atomicAdd(&deg[dst[e]], 1.0f);
}

__global__ void gcn_rsqrt_kernel(float* __restrict__ deg, int n) {
    int i = blockIdx.x * blockDim.x + threadIdx.x;
    if (i < n) {
        float d = deg[i];
        deg[i] = (d > 0.0f) ? rsqrtf(d) : 0.0f;
    }
}

// ---------------------------------------------------------------------------
// Dense GEMM  H[n_rows x NOUT] = relu?(X[n_rows x K]) @ W[K x NOUT]
//
// Block = 256 threads = 8 waves. blockIdx.y picks a 16-wide N panel of W,
// staged once into LDS (interleaved [(k/2)][n][2] so the per-step B fragment
// {W[k][n], W[k+1][n]} is a single aligned ds_load_b64). Each wave register-
// blocks 4 consecutive 16-row M tiles against that shared B fragment:
// per k-step = 1 ds_load_b64 + 4 global_load_b64 + 4 v_wmma_f32_16x16x4_f32.
//
// Fragment layouts per CDNA5 ISA 7.12.2 (f32):
//   A (16x4): lane m = lane&15, elem j holds K = k0 + 2*(lane>>4) + j
//   B (4x16): lane n = lane&15, elem j holds K = k0 + 2*(lane>>4) + j
//   C/D v8f : elem r -> row m0 + r + 8*(lane>>4), col n0 + (lane&15)
// All tile-validity branches are wave-uniform, so EXEC is all-ones at
// every WMMA as required.
// ---------------------------------------------------------------------------
template <int K, int NOUT, bool RELU>
__global__ void gcn_gemm_wmma_f32_kernel(const float* __restrict__ X,
                                         const float* __restrict__ W,
                                         float* __restrict__ H,
                                         int n_rows) {
    __shared__ float wp[K * 16];  // interleaved: ((k>>1)*16 + n)*2 + (k&1)

    const int n0 = blockIdx.y << 4;

    // Cooperative stage of W[:, n0:n0+16] into LDS
    for (int t = threadIdx.x; t < K * 16; t += blockDim.x) {
        int kk = t >> 4;
        int nn = t & 15;
        wp[((kk >> 1) * 16 + nn) * 2 + (kk & 1)] = W[(size_t)kk * NOUT + n0 + nn];
    }
    __syncthreads();

    const int lane  = threadIdx.x & 31;
    const int wid   = threadIdx.x >> 5;  // 0..7
    const int idx   = lane & 15;
    const int khalf = lane >> 4;

    const int tilesM = n_rows >> 4;
    const int mt0 = (blockIdx.x * 8 + wid) * 4;  // first 16-row tile of this wave
    if (mt0 >= tilesM) return;                   // wave-uniform exit

    bool valid[4];
    const float* xbase[4];
#pragma unroll
    for (int i = 0; i < 4; ++i) {
        int mt = mt0 + i;
        valid[i] = (mt < tilesM);
        int row = valid[i] ? ((mt << 4) + idx) : idx;  // dummy row 0..15 if invalid
        xbase[i] = X + (size_t)row * K + 2 * khalf;
    }

    v8f acc[4] = {};

#pragma unroll 4
    for (int k = 0; k < K; k += 4) {
        const int kb = k + 2 * khalf;  // even
        v2f b = *(const v2f*)&wp[((kb >> 1) * 16 + idx) * 2];
#pragma unroll
        for (int i = 0; i < 4; ++i) {
            v2f a = *(const v2f*)(xbase[i] + k);
            if (RELU) {
                a.x = fmaxf(a.x, 0.0f);
                a.y = fmaxf(a.y, 0.0f);
            }
            acc[i] = __builtin_amdgcn_wmma_f32_16x16x4_f32(
                         /*neg_a=*/false, a, /*neg_b=*/false, b,
                         /*c_mod=*/(short)0, acc[i],
                         /*reuse_a=*/false, /*reuse_b=*/false);
        }
    }

#pragma unroll
    for (int i = 0; i < 4; ++i) {
        if (!valid[i]) continue;  // wave-uniform
        int m0 = (mt0 + i) << 4;
        float* hrow = H + (size_t)(m0 + 8 * khalf) * NOUT + n0 + idx;
#pragma unroll
        for (int r = 0; r < 8; ++r) hrow[(size_t)r * NOUT] = acc[i][r];
    }
}

// ---------------------------------------------------------------------------
// agg[n][f] = bias[f] + h[n][f] * dis[n]^2     (bias + self-loop term)
// ---------------------------------------------------------------------------
__global__ void gcn_agg_init_kernel(const float* __restrict__ h,
                                    const float* __restrict__ bias,
                                    const float* __restrict__ dis,
                                    float* __restrict__ agg,
                                    int n_nodes, int log2F) {
    long long t = (long long)blockIdx.x * blockDim.x + threadIdx.x;
    long long total = (long long)n_nodes << log2F;
    if (t >= total) return;
    int n = (int)(t >> log2F);
    int f = (int)t & ((1 << log2F) - 1);
    float di = dis[n];
    agg[t] = bias[f] + h[t] * di * di;
}

// ---------------------------------------------------------------------------
// agg[dst][f] += h[src][f] * dis[src]*dis[dst], f32 atomics, float4 gather.
// Each thread handles one 4-feature chunk of one edge.
// ---------------------------------------------------------------------------
__global__ void gcn_agg_edges_kernel(const float4* __restrict__ h4,
                                     const int* __restrict__ src,
                                     const int* __restrict__ dst,
                                     const float* __restrict__ dis,
                                     float* __restrict__ agg,
                                     int nedges, int log2F) {
    const int cshift = log2F - 2;  // chunks per edge = F/4
    long long t = (long long)blockIdx.x * blockDim.x + threadIdx.x;
    long long e = t >> cshift;
    if (e >= nedges) return;
    int fc = (int)t & ((1 << cshift) - 1);

    int s = src[e];
    int d = dst[e];
    float w = dis[s] * dis[d];
    float4 v = h4[((size_t)s << cshift) + fc];
    float* o = agg + ((size_t)d << log2F) + ((size_t)fc << 2);
    atomicAdd(o + 0, v.x * w);
    atomicAdd(o + 1, v.y * w);
    atomicAdd(o + 2, v.z * w);
    atomicAdd(o + 3, v.w * w);
}

// ---------------------------------------------------------------------------
// out[n][0:7] = h3[n][0:16] @ Wl[16x7] + bl    (no relu on h3)
// ---------------------------------------------------------------------------
__global__ void gcn_final_linear_kernel(const float* __restrict__ h,
                                        const float* __restrict__ Wl,
                                        const float* __restrict__ bl,
                                        float* __restrict__ out, int n) {
    int i = blockIdx.x * blockDim.x + threadIdx.x;
    if (i >= n) return;
    float acc[7];
#pragma unroll
    for (int j = 0; j < 7; ++j) acc[j] = bl[j];
#pragma unroll
    for (int k = 0; k < 16; ++k) {
        float v = h[(size_t)i * 16 + k];
#pragma unroll
        for (int j = 0; j < 7; ++j) acc[j] = fmaf(v, Wl[k * 7 + j], acc[j]);
    }
#pragma unroll
    for (int j = 0; j < 7; ++j) out[(size_t)i * 7 + j] = acc[j];
}

// ---------------------------------------------------------------------------
// Host-side launch
// ---------------------------------------------------------------------------
extern "C" void kernel_launch(void* const* d_in, const int* in_sizes, int n_in,
                              void* d_out, int out_size, void* d_ws, size_t ws_size,
                              hipStream_t stream) {
    const float* x  = (const float*)d_in[0];
    const int*   ei = (const int*)d_in[1];  // [2, N_EDGES]
    const float* W1 = (const float*)d_in[2];
    const float* b1 = (const float*)d_in[3];
    const float* W2 = (const float*)d_in[4];
    const float* b2 = (const float*)d_in[5];
    const float* W3 = (const float*)d_in[6];
    const float* b3 = (const float*)d_in[7];
    const float* Wl = (const float*)d_in[8];
    const float* bl = (const float*)d_in[9];
    float* out = (float*)d_out;

    const int* src = ei;
    const int* dst = ei + GCN_N_EDGES;

    // Workspace carve-out: dis (400 KB) | bufA (25.6 MB) | bufB (25.6 MB)
    char* ws = (char*)d_ws;
    float* dis  = (float*)ws;
    float* bufA = (float*)(ws + (size_t)(1u << 19));                       // +512 KB
    float* bufB = (float*)(ws + (size_t)(1u << 19) + (size_t)(26u << 20)); // +26 MB

    const int TPB = 256;
    const int N = GCN_N_NODES;
    const int E = GCN_N_EDGES;

    const int tilesM  = N >> 4;                    // 6250
    const int mBlocks = (tilesM + 31) / 32;        // 8 waves * 4 tiles per block

    // --- normalization coefficients: dis = rsqrt(deg) with self-loops -----
    gcn_init_deg_kernel<<<(N + TPB - 1) / TPB, TPB, 0, stream>>>(dis, N);
    gcn_edge_deg_kernel<<<(E + TPB - 1) / TPB, TPB, 0, stream>>>(dst, dis, E);
    gcn_rsqrt_kernel<<<(N + TPB - 1) / TPB, TPB, 0, stream>>>(dis, N);

    // --- layer 1: 512 -> 64 ------------------------------------------------
    gcn_gemm_wmma_f32_kernel<512, 64, false>
        <<<dim3(mBlocks, 4), TPB, 0, stream>>>(x, W1, bufA, N);
    {
        long long tot = (long long)N << 6;
        gcn_agg_init_kernel<<<(int)((tot + TPB - 1) / TPB), TPB, 0, stream>>>(
            bufA, b1, dis, bufB, N, 6);
        long long et = (long long)E << 4;  // F/4 = 16 chunks per edge
        gcn_agg_edges_kernel<<<(int)((et + TPB - 1) / TPB), TPB, 0, stream>>>(
            (const float4*)bufA, src, dst, dis, bufB, E, 6);
    }

    // --- layer 2: 64 -> 32 (relu fused into GEMM input) --------------------
    gcn_gemm_wmma_f32_kernel<64, 32, true>
        <<<dim3(mBlocks, 2), TPB, 0, stream>>>(bufB, W2, bufA, N);
    {
        long long tot = (long long)N << 5;
        gcn_agg_init_kernel<<<(int)((tot + TPB - 1) / TPB), TPB, 0, stream>>>(
            bufA, b2, dis, bufB, N, 5);
        long long et = (long long)E << 3;  // F/4 = 8 chunks per edge
        gcn_agg_edges_kernel<<<(int)((et + TPB - 1) / TPB), TPB, 0, stream>>>(
            (const float4*)bufA, src, dst, dis, bufB, E, 5);
    }

    // --- layer 3: 32 -> 16 (relu fused into GEMM input) --------------------
    gcn_gemm_wmma_f32_kernel<32, 16, true>
        <<<dim3(mBlocks, 1), TPB, 0, stream>>>(bufB, W3, bufA, N);
    {
        long long tot = (long long)N << 4;
        gcn_agg_init_kernel<<<(int)((tot + TPB - 1) / TPB), TPB, 0, stream>>>(
            bufA, b3, dis, bufB, N, 4);
        long long et = (long long)E << 2;  // F/4 = 4 chunks per edge
        gcn_agg_edges_kernel<<<(int)((et + TPB - 1) / TPB), TPB, 0, stream>>>(
            (const float4*)bufA, src, dst, dis, bufB, E, 4);
    }

    // --- final linear: 16 -> 7 ---------------------------------------------
    gcn_final_linear_kernel<<<(N + TPB - 1) / TPB, TPB, 0, stream>>>(
        bufB, Wl, bl, out, N);
}